// RelativeGridAttention_65189013619044
// MI455X (gfx1250) — compile-verified
//
#include <hip/hip_runtime.h>

typedef __attribute__((ext_vector_type(2))) float v2f;
typedef __attribute__((ext_vector_type(4))) float v4f;
typedef __attribute__((ext_vector_type(8))) float v8f;

#define BATCH 2
#define HEADS 8
#define NQ    4096
#define DIM   32
#define GRIDH 32
#define GRIDW 32
#define KV    (GRIDH * GRIDW)     // 1024 keys per (b,h)
#define QTILE 16
#define LDS_STRIDE (KV + 16)      // 4160 B row skew -> conflict-free, 16B aligned

__global__ __launch_bounds__(256)
void rga_kernel(const float* __restrict__ Qg,
                const float* __restrict__ Kg,
                const int*   __restrict__ Pos,
                const float* __restrict__ Bias,
                float* __restrict__ Out)
{
    __shared__ float sc[QTILE * LDS_STRIDE];   // 16 x 1040 floats = 66.5 KB

    const int tid  = threadIdx.x;
    const int wave = tid >> 5;
    const int lane = tid & 31;
    const int col  = lane & 15;     // M (A) / N (B,C) index within 16
    const int hi   = lane >> 4;     // half-wave select per WMMA f32 layouts

    int idx = blockIdx.x;
    const int qt = idx & 255;       // NQ/QTILE = 256 tiles
    idx >>= 8;
    const int h  = idx & 7;
    const int b  = idx >> 3;

    const int qbase = qt * QTILE;
    const float* qptr = Qg + ((size_t)(b * HEADS + h) * NQ + qbase) * DIM;
    const float* kptr = Kg + (size_t)(b * HEADS + h) * KV * DIM;

    // ---------------- Phase 1: scores = Q x K^T via V_WMMA_F32_16X16X4_F32 ----
    // A (16x4 f32) fragment per 4-wide k-chunk: lanes 0-15 row=col hold K=kk*4+{0,1},
    // lanes 16-31 hold K=kk*4+{2,3}  (ISA 7.12.2 32-bit A layout).
    v2f a[8];
    #pragma unroll
    for (int kk = 0; kk < 8; ++kk) {
        const float* p = qptr + col * DIM + kk * 4 + 2 * hi;
        a[kk].x = p[0];
        a[kk].y = p[1];
    }

    #pragma unroll 1
    for (int t = 0; t < 8; ++t) {
        const int nbase = (wave * 8 + t) * 16;   // this wave's 16-key column tile
        v8f c = {};
        #pragma unroll
        for (int kk = 0; kk < 8; ++kk) {
            // B (4x16 f32): lanes 0-15 hold K=kk*4+{0,1}, lanes 16-31 K=kk*4+{2,3},
            // column N = col ; B[k][n] = keys[n][k]
            const float* p = kptr + (nbase + col) * DIM + kk * 4 + 2 * hi;
            v2f bf;
            bf.x = p[0];
            bf.y = p[1];
            c = __builtin_amdgcn_wmma_f32_16x16x4_f32(false, a[kk], false, bf,
                                                      (short)0, c, false, false);
        }
        // C/D 16x16 f32 layout: c[r] holds (M = r + 8*hi, N = col)
        #pragma unroll
        for (int r = 0; r < 8; ++r)
            sc[(r + 8 * hi) * LDS_STRIDE + nbase + col] = c[r];
    }

    __syncthreads();

    // ---------------- Phase 2: bias gather + rowwise softmax (vectorized) ----
    // 16 threads per query row; each thread owns 16 chunks of 4 contiguous cols.
    // A 4-aligned chunk never crosses a 32-column grid-row boundary.
    const int row = tid >> 4;
    const int sub = tid & 15;
    const int q   = qbase + row;

    const int py = Pos[(b * NQ + q) * 2 + 0];
    const int px = Pos[(b * NQ + q) * 2 + 1];
    const float* brow = Bias + (size_t)h * (2 * GRIDH - 1) * (2 * GRIDW - 1);
    const int bconst = (GRIDH - 1 - py) * (2 * GRIDW - 1) + (GRIDW - 1 - px);
    const float scale = 0.17677669529663687f;   // 1/sqrt(32)

    v4f* myrow4 = (v4f*)(sc + row * LDS_STRIDE);

    float m = -3.0e38f;
    #pragma unroll 4
    for (int k = 0; k < KV / 64; ++k) {
        const int n4 = sub + 16 * k;            // float4 chunk index
        const int n  = 4 * n4;                  // first column of chunk
        const int i  = n >> 5;                  // grid row (same for all 4)
        const int j  = n & 31;
        const float* bp = brow + i * (2 * GRIDW - 1) + j + bconst;
        v4f l = myrow4[n4];
        l.x = l.x * scale + bp[0];
        l.y = l.y * scale + bp[1];
        l.z = l.z * scale + bp[2];
        l.w = l.w * scale + bp[3];
        myrow4[n4] = l;
        m = fmaxf(m, fmaxf(fmaxf(l.x, l.y), fmaxf(l.z, l.w)));
    }
    m = fmaxf(m, __shfl_xor(m, 1));
    m = fmaxf(m, __shfl_xor(m, 2));
    m = fmaxf(m, __shfl_xor(m, 4));
    m = fmaxf(m, __shfl_xor(m, 8));

    float s = 0.0f;
    #pragma unroll 4
    for (int k = 0; k < KV / 64; ++k) {
        const int n4 = sub + 16 * k;
        v4f l = myrow4[n4];
        v4f e;
        e.x = __expf(l.x - m);
        e.y = __expf(l.y - m);
        e.z = __expf(l.z - m);
        e.w = __expf(l.w - m);
        myrow4[n4] = e;
        s += (e.x + e.y) + (e.z + e.w);
    }
    s += __shfl_xor(s, 1);
    s += __shfl_xor(s, 2);
    s += __shfl_xor(s, 4);
    s += __shfl_xor(s, 8);
    const float inv = 1.0f / s;

    v4f* orow4 = (v4f*)(Out + ((size_t)(b * HEADS + h) * NQ + q) * KV);
    #pragma unroll 4
    for (int k = 0; k < KV / 64; ++k) {
        const int n4 = sub + 16 * k;
        v4f e = myrow4[n4];
        e.x *= inv; e.y *= inv; e.z *= inv; e.w *= inv;
        __builtin_nontemporal_store(e, orow4 + n4);   // global_store_b128, TH=NT
    }
}

extern "C" void kernel_launch(void* const* d_in, const int* in_sizes, int n_in,
                              void* d_out, int out_size, void* d_ws, size_t ws_size,
                              hipStream_t stream) {
    const float* q   = (const float*)d_in[0];
    const float* k   = (const float*)d_in[1];
    const int*   pos = (const int*)d_in[2];
    const float* rb  = (const float*)d_in[3];
    float* out = (float*)d_out;

    dim3 grid(BATCH * HEADS * (NQ / QTILE));   // 4096 workgroups
    dim3 block(256);                            // 8 wave32s
    rga_kernel<<<grid, block, 0, stream>>>(q, k, pos, rb, out);
}